// CentroidDistance_80668075754257
// MI455X (gfx1250) — compile-verified
//
#include <hip/hip_runtime.h>
#include <hip/hip_bf16.h>

// CDNA5 / gfx1250, wave32. Uses V_WMMA_F32_16X16X4_F32 (full fp32 matrix path).

typedef float v2f __attribute__((ext_vector_type(2)));
typedef float v8f __attribute__((ext_vector_type(8)));

#define EMBED      128
#define K_CENT     128
#define POIN_EPS   1e-6f
#define LDS_STRIDE 132   // 128 + 4 pad: B-lane addresses land on distinct LDS banks

// ---------------- workspace layout (floats) ----------------
// W[0]        : sum(mask) accumulator
// W[8..135]   : c2[k] = ||centroid_k||^2
// W[256..383] : graph-level distance sum accumulator per centroid
// ------------------------------------------------------------

__global__ __launch_bounds__(256)
void cd_init_kernel(const float* __restrict__ cent, float* __restrict__ W) {
    int tid = threadIdx.x;
    if (tid == 0) W[0] = 0.f;
    if (tid < K_CENT) {
        W[256 + tid] = 0.f;
        const float* row = cent + tid * EMBED;
        float s = 0.f;
        #pragma unroll 8
        for (int d = 0; d < EMBED; ++d) s += row[d] * row[d];
        W[8 + tid] = s;
    }
}

__global__ __launch_bounds__(256)
void cd_masksum_kernel(const float* __restrict__ mask, int n, float* __restrict__ W) {
    __shared__ float s[256];
    int tid = threadIdx.x;
    float v = 0.f;
    for (int i = blockIdx.x * 256 + tid; i < n; i += gridDim.x * 256) v += mask[i];
    s[tid] = v;
    __syncthreads();
    #pragma unroll
    for (int off = 128; off > 0; off >>= 1) {
        if (tid < off) s[tid] += s[tid + off];
        __syncthreads();
    }
    if (tid == 0) atomicAdd(&W[0], s[0]);
}

// Main kernel: each workgroup = 8 wave32s; wave w computes a 16-node x 128-centroid
// tile of dot products via v_wmma_f32_16x16x4_f32, then the Poincare epilogue.
__global__ __launch_bounds__(256)
void cd_poincare_wmma_kernel(const float* __restrict__ x,
                             const float* __restrict__ mask,
                             const float* __restrict__ cent,
                             float* __restrict__ W,
                             float* __restrict__ out,
                             int n) {
    __shared__ float sC[K_CENT * LDS_STRIDE];  // centroids, padded row stride
    __shared__ float sX2[8 * 32];              // per-lane ||x||^2 partials
    __shared__ float sG[K_CENT];               // per-workgroup graph-sum

    const int tid = threadIdx.x;

    // Cooperative load of the 128x128 f32 centroid matrix into LDS (float4 units).
    {
        const float4* src = (const float4*)cent;
        for (int u = tid; u < K_CENT * (EMBED / 4); u += 256) {
            int r  = u >> 5;       // centroid row
            int c4 = u & 31;       // float4 column
            float4 v = src[r * (EMBED / 4) + c4];
            float* dst = &sC[r * LDS_STRIDE + c4 * 4];
            dst[0] = v.x; dst[1] = v.y; dst[2] = v.z; dst[3] = v.w;
        }
    }
    if (tid < K_CENT) sG[tid] = 0.f;
    __syncthreads();

    const int wave = tid >> 5;
    const int lane = tid & 31;
    const int lo   = lane & 15;   // M (A) / N (B) index within the 16-wide tile
    const int hi   = lane >> 4;   // selects K-pair {0,1} vs {2,3} within a K-step

    const long tileBase = (long)blockIdx.x * 128 + (long)wave * 16;

    // A-operand row for this lane; clamp for the grid tail so EXEC stays all-ones.
    long rowA = tileBase + lo;
    if (rowA > (long)n - 1) rowA = (long)n - 1;
    const v2f* aRow = (const v2f*)(x + rowA * (long)EMBED);

    v8f acc[8] = {};
    float sum2 = 0.f;

    // 32 K-steps of 4: 256 wmma per wave-tile (16*128*128*2 = 524288 FLOP).
    #pragma unroll 4
    for (int k0 = 0; k0 < EMBED; k0 += 4) {
        // A 16x4 f32 layout: lanes 0-15 hold (k0,k0+1), lanes 16-31 hold (k0+2,k0+3)
        v2f a = aRow[(k0 >> 1) + hi];
        sum2 += a.x * a.x + a.y * a.y;   // free ||x||^2 partial from the A stream
        #pragma unroll
        for (int t = 0; t < 8; ++t) {
            // B 4x16 f32: B[k][col] = c[col][k], col = t*16 + lo
            const v2f b = *(const v2f*)&sC[(t * 16 + lo) * LDS_STRIDE + k0 + 2 * hi];
            acc[t] = __builtin_amdgcn_wmma_f32_16x16x4_f32(
                false, a, false, b, (short)0, acc[t], false, false);
        }
    }

    // Combine the two half-lane ||x||^2 partials per row via LDS.
    sX2[wave * 32 + lane] = sum2;
    __syncthreads();

    // Per-row epilogue constants (C/D layout: VGPR i, lane -> row = i + 8*hi).
    float x2v[8], mv[8];
    #pragma unroll
    for (int i = 0; i < 8; ++i) {
        int  rloc = i + 8 * hi;
        long row  = tileBase + rloc;
        x2v[i] = sX2[wave * 32 + rloc] + sX2[wave * 32 + 16 + rloc];
        mv[i]  = (row < n) ? mask[row] : 0.f;
    }

    #pragma unroll
    for (int t = 0; t < 8; ++t) {
        const int col = t * 16 + lo;
        const float c2 = W[8 + col];
        float gpart = 0.f;
        #pragma unroll
        for (int i = 0; i < 8; ++i) {
            long row  = tileBase + i + 8 * hi;
            float dot = acc[t][i];
            float x2  = x2v[i];
            float sq    = fmaxf(x2 + c2 - 2.f * dot, 0.f);          // ||x-c||^2
            float denom = fmaxf((1.f - x2) * (1.f - c2), POIN_EPS);
            float z     = fmaxf(1.f + 2.f * sq / denom, 1.f + POIN_EPS);
            float dist  = logf(z + sqrtf(fmaxf(z * z - 1.f, 0.f))); // arccosh
            float o     = dist * mv[i];
            if (row < n) out[128 + row * (long)K_CENT + col] = o;
            gpart += o;   // mv==0 for clamped tail rows -> no pollution
        }
        atomicAdd(&sG[col], gpart);   // ds_add_f32, workgroup-local
    }
    __syncthreads();
    if (tid < K_CENT) atomicAdd(&W[256 + tid], sG[tid]);   // one global atomic per col
}

__global__ __launch_bounds__(128)
void cd_finalize_kernel(const float* __restrict__ W, float* __restrict__ out) {
    int k = threadIdx.x;
    if (k < K_CENT) out[k] = W[256 + k] / W[0];
}

extern "C" void kernel_launch(void* const* d_in, const int* in_sizes, int n_in,
                              void* d_out, int out_size, void* d_ws, size_t ws_size,
                              hipStream_t stream) {
    const float* x    = (const float*)d_in[0];   // node_repr [N,128]
    const float* mask = (const float*)d_in[1];   // mask      [N,1]
    const float* cent = (const float*)d_in[2];   // centroid  [128,128]
    float* out = (float*)d_out;                  // [128] graph dist ++ [N*128] node dist
    float* W   = (float*)d_ws;                   // >= 384 floats used
    const int n = in_sizes[1];                   // N (mask element count)

    cd_init_kernel<<<1, 256, 0, stream>>>(cent, W);
    cd_masksum_kernel<<<256, 256, 0, stream>>>(mask, n, W);
    const int blocks = (n + 127) / 128;
    cd_poincare_wmma_kernel<<<blocks, 256, 0, stream>>>(x, mask, cent, W, out, n);
    cd_finalize_kernel<<<1, 128, 0, stream>>>(W, out);
}